// Net_14113262535138
// MI455X (gfx1250) — compile-verified
//
#include <hip/hip_runtime.h>

// ---------------------------------------------------------------------------
// GraphSAGE 2-layer + global mean pool for MI455X (gfx1250, wave32).
// Memory-bound (~600 MB edge traffic @ 23.3 TB/s HBM; intermediates fit in
// the 192 MB L2). Dense node transforms use V_WMMA_F32_16X16X4_F32 on
// 16-node tiles; WMMA outputs are bounced through LDS so h1 is written with
// coalesced global_store_b128 instead of predicated dword scatters.
// ---------------------------------------------------------------------------

typedef __attribute__((ext_vector_type(2))) float v2f;
typedef __attribute__((ext_vector_type(8))) float v8f;

#define IN_F 8
#define H1   16
#define H2   20

// ---------------------------------------------------------------- zero scratch
__global__ void zero_kernel(float* __restrict__ p, long long n) {
    long long i = (long long)blockIdx.x * blockDim.x + threadIdx.x;
    if (i < n) p[i] = 0.0f;
}

// -------------------------------------------------- edge scatter: layer-1 agg
__global__ void scatter1_kernel(const float* __restrict__ x,
                                const int* __restrict__ ei, int E,
                                float* __restrict__ agg1,
                                float* __restrict__ cnt) {
    int e = blockIdx.x * blockDim.x + threadIdx.x;
    if (e >= E) return;
    int s = ei[e];          // src row
    int d = ei[E + e];      // dst row
    const float4* xr = (const float4*)(x + (size_t)s * IN_F);
    float* o = agg1 + (size_t)d * IN_F;
    __builtin_prefetch(o, 1, 3);            // pull dst line toward L2 (RMW soon)
    float4 a = xr[0], b = xr[1];
    atomicAdd(o + 0, a.x); atomicAdd(o + 1, a.y);
    atomicAdd(o + 2, a.z); atomicAdd(o + 3, a.w);
    atomicAdd(o + 4, b.x); atomicAdd(o + 5, b.y);
    atomicAdd(o + 6, b.z); atomicAdd(o + 7, b.w);
    atomicAdd(cnt + d, 1.0f);
}

// -------------------------------------------------- edge scatter: layer-2 agg
__global__ void scatter2_kernel(const float* __restrict__ h1,
                                const int* __restrict__ ei, int E,
                                float* __restrict__ agg2) {
    int e = blockIdx.x * blockDim.x + threadIdx.x;
    if (e >= E) return;
    int s = ei[e];
    int d = ei[E + e];
    const float4* hr = (const float4*)(h1 + (size_t)s * H1);
    float* o = agg2 + (size_t)d * H1;
    __builtin_prefetch(o, 1, 3);
#pragma unroll
    for (int q = 0; q < 4; ++q) {
        float4 v = hr[q];
        atomicAdd(o + q * 4 + 0, v.x); atomicAdd(o + q * 4 + 1, v.y);
        atomicAdd(o + q * 4 + 2, v.z); atomicAdd(o + q * 4 + 3, v.w);
    }
}

// ----------------------------------------------------------- pool node counts
__global__ void pool_count_kernel(const int* __restrict__ batch, int N,
                                  float* __restrict__ pcnt) {
    int n = blockIdx.x * blockDim.x + threadIdx.x;
    if (n < N) atomicAdd(&pcnt[batch[n]], 1.0f);
}

// ---------------------------------------------------------------------------
// Layer 1: h1 = relu([x | mean1] @ [W1r ; W1l] + b1), 16-node tile per wave.
// A: 16x16 f32, B: 16x16 f32 -> 4 chained V_WMMA_F32_16X16X4_F32.
// Fragment layouts per CDNA5 ISA 7.12.2 (f32, wave32):
//   A: lane(0..15)=M rows K={kb,kb+1}; lanes 16..31 same M, K={kb+2,kb+3}
//   B: VGPR0/1 rows K striped across lanes (N = lane&15, +2 rows for hi half)
//   C/D: VGPR r -> M = r + 8*(lane>=16), N = lane&15
// Output is staged back through the per-wave LDS tile so global stores are
// coalesced b128 with a single bounds check per lane.
// ---------------------------------------------------------------------------
__global__ void __launch_bounds__(256)
layer1_kernel(const float* __restrict__ x,
              const float* __restrict__ agg1,
              const float* __restrict__ cnt,
              const float* __restrict__ W1l,
              const float* __restrict__ b1,
              const float* __restrict__ W1r,
              float* __restrict__ h1, int N) {
    __shared__ float Af[8][16][16];   // per-wave A tile, reused for D tile
    __shared__ float Bw[16][16];      // shared B tile (concat weights)

    const int tid  = threadIdx.x;
    const int w    = tid >> 5;
    const int lane = tid & 31;
    const int half = lane >> 4;
    const int row  = lane & 15;       // M for A staging / frags
    const int col  = lane & 15;       // N for B frags and C/D
    const int n0   = (blockIdx.x * 8 + w) * 16;
    const int node = n0 + row;
    const int nodeC = node < N ? node : (N - 1);

    // stage B = [W1r ; W1l]  (A cols 0..7 = x -> W1r rows; cols 8..15 = mean)
    {
        int k = tid >> 4, j = tid & 15;
        Bw[k][j] = (k < IN_F) ? W1r[k * H1 + j] : W1l[(k - IN_F) * H1 + j];
    }
    // stage A rows: half 0 -> x features, half 1 -> mean = agg1 / max(cnt,1)
    {
        float cv  = cnt[nodeC];
        float inv = 1.0f / fmaxf(cv, 1.0f);
        float s   = half ? inv : 1.0f;
        const float* src = half ? (agg1 + (size_t)nodeC * IN_F)
                                : (x    + (size_t)nodeC * IN_F);
        const float4* sv = (const float4*)src;
        float4 f0 = sv[0], f1 = sv[1];
        float* dst = &Af[w][row][half * 8];
        dst[0] = f0.x * s; dst[1] = f0.y * s; dst[2] = f0.z * s; dst[3] = f0.w * s;
        dst[4] = f1.x * s; dst[5] = f1.y * s; dst[6] = f1.z * s; dst[7] = f1.w * s;
    }
    __syncthreads();

    v8f c = {};
#pragma unroll
    for (int kc = 0; kc < 4; ++kc) {
        const int kb = kc * 4 + half * 2;
        v2f a, b;
        a.x = Af[w][row][kb];  a.y = Af[w][row][kb + 1];
        b.x = Bw[kb][col];     b.y = Bw[kb + 1][col];
        c = __builtin_amdgcn_wmma_f32_16x16x4_f32(false, a, false, b,
                                                  (short)0, c, false, false);
    }

    // bias + ReLU, bounce D tile through LDS (A tile is dead now)
    const float bias = b1[col];
#pragma unroll
    for (int r = 0; r < 8; ++r) {
        float v = c[r] + bias;
        Af[w][r + half * 8][col] = v > 0.0f ? v : 0.0f;
    }
    __syncthreads();

    // coalesced write-out: each lane stores half a node row (8 floats, 2xb128)
    {
        const int orow  = lane & 15;
        const int ohalf = half * 8;
        const int onode = n0 + orow;
        if (onode < N) {
            const float4* sv = (const float4*)&Af[w][orow][ohalf];
            float4* dv = (float4*)(h1 + (size_t)onode * H1 + ohalf);
            dv[0] = sv[0];
            dv[1] = sv[1];
        }
    }
}

// ---------------------------------------------------------------------------
// Layer 2: h2 = leaky([h1 | mean2] @ [W2r ; W2l] + b2), fused with mean-pool
// scatter. A: 16x32, B padded to 32x32 (two 16-col N tiles, cols 20..31 zero).
// 8 K-chunks x 2 N-tiles = 16 WMMAs per wave. h2 never touches memory: it is
// accumulated straight into the pooled sums.
// ---------------------------------------------------------------------------
__global__ void __launch_bounds__(256)
layer2_kernel(const float* __restrict__ h1,
              const float* __restrict__ agg2,
              const float* __restrict__ cnt,
              const float* __restrict__ W2l,
              const float* __restrict__ b2,
              const float* __restrict__ W2r,
              const int* __restrict__ batch,
              float* __restrict__ psum, int N) {
    __shared__ float Af[8][16][32];
    __shared__ float Bw[32][32];

    const int tid  = threadIdx.x;
    const int w    = tid >> 5;
    const int lane = tid & 31;
    const int half = lane >> 4;
    const int row  = lane & 15;
    const int col  = lane & 15;
    const int n0   = (blockIdx.x * 8 + w) * 16;
    const int node = n0 + row;
    const int nodeC = node < N ? node : (N - 1);

    // stage B = [W2r ; W2l], N padded 20 -> 32
    for (int t = tid; t < 32 * 32; t += 256) {
        int k = t >> 5, j = t & 31;
        float v = 0.0f;
        if (j < H2) v = (k < H1) ? W2r[k * H2 + j] : W2l[(k - H1) * H2 + j];
        Bw[k][j] = v;
    }
    // stage A rows: half 0 -> h1 row, half 1 -> mean2 = agg2 / max(cnt,1)
    {
        float cv  = cnt[nodeC];
        float inv = 1.0f / fmaxf(cv, 1.0f);
        float s   = half ? inv : 1.0f;
        const float* src = half ? (agg2 + (size_t)nodeC * H1)
                                : (h1   + (size_t)nodeC * H1);
        const float4* sv = (const float4*)src;
        float* dst = &Af[w][row][half * 16];
#pragma unroll
        for (int q = 0; q < 4; ++q) {
            float4 f = sv[q];
            dst[q * 4 + 0] = f.x * s; dst[q * 4 + 1] = f.y * s;
            dst[q * 4 + 2] = f.z * s; dst[q * 4 + 3] = f.w * s;
        }
    }
    __syncthreads();

    v8f c0 = {}, c1 = {};
#pragma unroll
    for (int kc = 0; kc < 8; ++kc) {
        const int kb = kc * 4 + half * 2;
        v2f a, bA, bB;
        a.x  = Af[w][row][kb];   a.y  = Af[w][row][kb + 1];
        bA.x = Bw[kb][col];      bA.y = Bw[kb + 1][col];
        bB.x = Bw[kb][16 + col]; bB.y = Bw[kb + 1][16 + col];
        c0 = __builtin_amdgcn_wmma_f32_16x16x4_f32(false, a, false, bA,
                                                   (short)0, c0, false, false);
        c1 = __builtin_amdgcn_wmma_f32_16x16x4_f32(false, a, false, bB,
                                                   (short)0, c1, false, false);
    }

    const float bias0 = b2[col];             // cols 0..15 always valid (<20)
    const int   j1    = 16 + col;            // cols 16..31, valid iff col<4
    const float bias1 = b2[j1 < H2 ? j1 : 0];

#pragma unroll
    for (int r = 0; r < 8; ++r) {
        int m  = r + half * 8;
        int nn = n0 + m;
        if (nn < N) {
            int g = batch[nn];
            float v0 = c0[r] + bias0;
            v0 = v0 > 0.0f ? v0 : 0.1f * v0;           // LeakyReLU(0.1)
            atomicAdd(&psum[(size_t)g * H2 + col], v0);
            if (j1 < H2) {
                float v1 = c1[r] + bias1;
                v1 = v1 > 0.0f ? v1 : 0.1f * v1;
                atomicAdd(&psum[(size_t)g * H2 + j1], v1);
            }
        }
    }
}

// ------------------------------------------------------------------ finalize
__global__ void finalize_kernel(const float* __restrict__ psum,
                                const float* __restrict__ pcnt,
                                float* __restrict__ out, int total) {
    int i = blockIdx.x * blockDim.x + threadIdx.x;
    if (i < total) {
        int g = i / H2;
        out[i] = psum[i] / fmaxf(pcnt[g], 1.0f);
    }
}

// ---------------------------------------------------------------------------
extern "C" void kernel_launch(void* const* d_in, const int* in_sizes, int n_in,
                              void* d_out, int out_size, void* d_ws, size_t ws_size,
                              hipStream_t stream) {
    const float* x    = (const float*)d_in[0];
    const int*   ei   = (const int*)  d_in[1];
    const int*   bat  = (const int*)  d_in[2];
    const float* W1l  = (const float*)d_in[3];
    const float* b1   = (const float*)d_in[4];
    const float* W1r  = (const float*)d_in[5];
    const float* W2l  = (const float*)d_in[6];
    const float* b2   = (const float*)d_in[7];
    const float* W2r  = (const float*)d_in[8];
    float*       out  = (float*)d_out;

    const int N = in_sizes[0] / IN_F;
    const int E = in_sizes[1] / 2;
    const int G = out_size / H2;

    // scratch layout (floats)
    float* ws = (float*)d_ws;
    size_t off = 0;
    float* agg1 = ws + off; off += (size_t)N * IN_F;   // layer-1 segment sums
    float* cnt  = ws + off; off += (size_t)N;          // per-node edge counts
    float* h1   = ws + off; off += (size_t)N * H1;     // layer-1 activations
    float* agg2 = ws + off; off += (size_t)N * H1;     // layer-2 segment sums
    float* psum = ws + off; off += (size_t)G * H2;     // pooled sums
    float* pcnt = ws + off; off += (size_t)G;          // nodes per graph
    const long long total_ws = (long long)off;

    const int T = 256;
    // 1) zero scratch (deterministic per call; harness does not re-poison)
    zero_kernel<<<(int)((total_ws + T - 1) / T), T, 0, stream>>>(ws, total_ws);
    // 2) layer-1 edge scatter (mean numerator + counts)
    scatter1_kernel<<<(E + T - 1) / T, T, 0, stream>>>(x, ei, E, agg1, cnt);
    // 3) layer-1 node transform via WMMA (16 nodes per wave, 8 waves per block)
    const int tiles  = (N + 15) / 16;
    const int blocks = (tiles + 7) / 8;
    layer1_kernel<<<blocks, T, 0, stream>>>(x, agg1, cnt, W1l, b1, W1r, h1, N);
    // 4) layer-2 edge scatter
    scatter2_kernel<<<(E + T - 1) / T, T, 0, stream>>>(h1, ei, E, agg2);
    // 5) pool counts
    pool_count_kernel<<<(N + T - 1) / T, T, 0, stream>>>(bat, N, pcnt);
    // 6) layer-2 node transform via WMMA, fused LeakyReLU + pool scatter
    layer2_kernel<<<blocks, T, 0, stream>>>(h1, agg2, cnt, W2l, b2, W2r,
                                            bat, psum, N);
    // 7) divide pooled sums by counts
    finalize_kernel<<<(G * H2 + T - 1) / T, T, 0, stream>>>(psum, pcnt, out,
                                                            G * H2);
}